// MultiScaleDeformableAttention_9371618640482
// MI455X (gfx1250) — compile-verified
//
#include <hip/hip_runtime.h>
#include <math.h>
#include <stdint.h>

typedef __attribute__((ext_vector_type(2))) float v2f;
typedef __attribute__((ext_vector_type(8))) float v8f;

#define N_B     2
#define LQ      21760
#define DM      256
#define NH      8
#define NLVL    4
#define NPTS    4
#define DH      32
#define LEN_IN  21760
#define ROWS    (N_B * LQ)          // 43520

#define GEMM_K    256               // all GEMMs in this model have K == 256
#define LDS_PITCH 258               // 256 + 2 pad: kills 16-way bank conflicts,
                                    // keeps row base 8B-aligned for ds_load_b64

// ---------------------------------------------------------------------------
// f32 WMMA GEMM:  C[M,N] = A[M,K=256] * B[256,N] + bias[N]
// Block = 128 threads (4 waves). The block's 16x256 A tile is staged into LDS
// once via GLOBAL_LOAD_ASYNC_TO_LDS_B128 (ASYNCcnt), then each wave computes a
// 16x64 C strip with V_WMMA_F32_16X16X4_F32, reading A fragments from LDS.
//
// Fragment layouts (CDNA5 ISA):
//   A 16x4 : lane M = lane%16; VGPR0 = K+(lane<16?0:2), VGPR1 = +1.
//   B 4x16 : lane N = lane%16; VGPR0 = row K+(lane<16?0:2), VGPR1 = +1.
//   C 16x16: VGPR i holds M = i + (lane<16?0:8), N = lane%16.
// ---------------------------------------------------------------------------
__global__ __launch_bounds__(128) void gemm_wmma_f32(
    const float* __restrict__ A, const float* __restrict__ B,
    const float* __restrict__ bias, float* __restrict__ C,
    int Mrows, int Ncols)
{
    __shared__ float As[16 * LDS_PITCH];   // 16 rows x 256 K (padded)

    const int tid  = threadIdx.x;
    const int lane = tid & 31;
    const int wave = tid >> 5;
    const int hlf  = lane >> 4;            // 0: lanes 0-15, 1: lanes 16-31
    const int l16  = lane & 15;

    const int rowBase = blockIdx.x * 16;
    if (rowBase >= Mrows) return;

    // ---- async stage of the 16x256 A tile: 1024 float4s, 8 per thread ----
#pragma unroll
    for (int j = 0; j < 8; ++j) {
        const int f    = tid + j * 128;    // float4 index 0..1023
        const int row  = f >> 6;           // 0..15
        const int kpos = (f & 63) * 4;     // 0..252
        const uint64_t gaddr =
            (uint64_t)(A + (size_t)(rowBase + row) * GEMM_K + kpos);
        const uint32_t laddr =
            (uint32_t)(size_t)(&As[row * LDS_PITCH + kpos]);
        asm volatile("global_load_async_to_lds_b128 %0, %1, off"
                     :: "v"(laddr), "v"(gaddr) : "memory");
    }
    asm volatile("s_wait_asynccnt 0x0" ::: "memory");
    __syncthreads();

    const int colBase = (blockIdx.y * (blockDim.x >> 5) + wave) * 64;
    if (colBase >= Ncols) return;

    const float* __restrict__ Alds = &As[l16 * LDS_PITCH];

    v8f acc[4];
    acc[0] = v8f{}; acc[1] = v8f{}; acc[2] = v8f{}; acc[3] = v8f{};

    for (int k = 0; k < GEMM_K; k += 4) {
        const int ka = k + hlf * 2;

        if ((k & 31) == 0 && k + 32 < GEMM_K)
            __builtin_prefetch(B + (size_t)(k + 32) * Ncols + colBase + l16, 0, 1);

        v2f a;                              // ds_load_b64 from staged tile
        a.x = Alds[ka];
        a.y = Alds[ka + 1];

        const float* __restrict__ B0 = B + (size_t)ka * Ncols + colBase + l16;
        const float* __restrict__ B1 = B0 + Ncols;

        v2f b0; b0.x = B0[0];  b0.y = B1[0];
        v2f b1; b1.x = B0[16]; b1.y = B1[16];
        v2f b2; b2.x = B0[32]; b2.y = B1[32];
        v2f b3; b3.x = B0[48]; b3.y = B1[48];

        acc[0] = __builtin_amdgcn_wmma_f32_16x16x4_f32(false, a, false, b0, (short)0, acc[0], false, false);
        acc[1] = __builtin_amdgcn_wmma_f32_16x16x4_f32(false, a, false, b1, (short)0, acc[1], false, false);
        acc[2] = __builtin_amdgcn_wmma_f32_16x16x4_f32(false, a, false, b2, (short)0, acc[2], false, false);
        acc[3] = __builtin_amdgcn_wmma_f32_16x16x4_f32(false, a, false, b3, (short)0, acc[3], false, false);
    }

#pragma unroll
    for (int t = 0; t < 4; ++t) {
        const int col = colBase + t * 16 + l16;
        const float bv = bias ? bias[col] : 0.0f;
#pragma unroll
        for (int i = 0; i < 8; ++i) {
            const int row = rowBase + hlf * 8 + i;
            C[(size_t)row * Ncols + col] = acc[t][i] + bv;
        }
    }
}

// ---------------------------------------------------------------------------
// Softmax over 16 logits per (n, q, head) row, in place.
// ---------------------------------------------------------------------------
__global__ __launch_bounds__(256) void softmax16_kernel(float* __restrict__ aw, int nrows)
{
    const int i = blockIdx.x * blockDim.x + threadIdx.x;
    if (i >= nrows) return;
    float* p = aw + (size_t)i * 16;

    float v[16];
    float mx = -3.4e38f;
#pragma unroll
    for (int j = 0; j < 16; ++j) { v[j] = p[j]; mx = fmaxf(mx, v[j]); }
    float s = 0.0f;
#pragma unroll
    for (int j = 0; j < 16; ++j) { v[j] = __expf(v[j] - mx); s += v[j]; }
    const float inv = 1.0f / s;
#pragma unroll
    for (int j = 0; j < 16; ++j) p[j] = v[j] * inv;
}

// ---------------------------------------------------------------------------
// Bilinear deformable sampling. One wave32 per (n, q, head); lane = channel.
// v layout: [N, LEN_IN, NH, DH]  (lanes of a wave read 128B coalesced)
// off:      [N, LQ, NH, NLVL, NPTS, 2]  (= the [ROWS,256] GEMM output)
// aw:       [N, LQ, NH, 16]             (softmaxed)
// heads:    [N, LQ, NH*DH] = [ROWS, 256]
// ---------------------------------------------------------------------------
__global__ __launch_bounds__(256) void msda_sample_kernel(
    const float* __restrict__ v, const float* __restrict__ off,
    const float* __restrict__ aw, const float* __restrict__ ref,
    float* __restrict__ heads)
{
    const int wid  = blockIdx.x * (blockDim.x >> 5) + (threadIdx.x >> 5);
    const int lane = threadIdx.x & 31;

    const int m = wid & (NH - 1);
    const int q = (wid >> 3) % LQ;
    const int n = wid / (NH * LQ);

    const size_t rowq = (size_t)n * LQ + q;

    // lane-parallel staging of per-(n,q,m) offsets (32 floats) and weights (16)
    const float offv = off[rowq * 256 + m * 32 + lane];
    const float awv  = aw[rowq * 128 + m * 16 + (lane & 15)];

    const int lvlOff[NLVL] = {0, 16384, 20480, 21504};
    const int lvlH[NLVL]   = {128, 64, 32, 16};
    const int lvlW[NLVL]   = {128, 64, 32, 16};

    // v indexed by spatial position: vbase[pos * 256]
    const float* __restrict__ vbase = v + (size_t)n * LEN_IN * 256 + m * 32 + lane;

    float acc = 0.0f;

#pragma unroll
    for (int l = 0; l < NLVL; ++l) {
        const int H = lvlH[l], W = lvlW[l];
        const float rx = ref[(rowq * NLVL + l) * 2 + 0];
        const float ry = ref[(rowq * NLVL + l) * 2 + 1];
        const float* __restrict__ vl = vbase + (size_t)lvlOff[l] * 256;

#pragma unroll
        for (int p = 0; p < NPTS; ++p) {
            const int idx = l * NPTS + p;
            const float ox = __shfl(offv, idx * 2 + 0, 32);
            const float oy = __shfl(offv, idx * 2 + 1, 32);
            const float w  = __shfl(awv,  idx, 32);

            // grid_sample(align_corners=False) algebra collapses to:
            const float gx = rx * (float)W + ox - 0.5f;
            const float gy = ry * (float)H + oy - 0.5f;

            const float x0f = floorf(gx), y0f = floorf(gy);
            const float wx1 = gx - x0f,  wy1 = gy - y0f;
            const float wx0 = 1.0f - wx1, wy0 = 1.0f - wy1;
            const int x0 = (int)x0f, y0 = (int)y0f;
            const int x1 = x0 + 1,   y1 = y0 + 1;

            float s = 0.0f;
            if (y0 >= 0 && y0 < H) {
                const float* r0 = vl + (size_t)y0 * W * 256;
                if (x0 >= 0 && x0 < W) s += wy0 * wx0 * r0[(size_t)x0 * 256];
                if (x1 >= 0 && x1 < W) s += wy0 * wx1 * r0[(size_t)x1 * 256];
            }
            if (y1 >= 0 && y1 < H) {
                const float* r1 = vl + (size_t)(y1) * W * 256;
                if (x0 >= 0 && x0 < W) s += wy1 * wx0 * r1[(size_t)x0 * 256];
                if (x1 >= 0 && x1 < W) s += wy1 * wx1 * r1[(size_t)x1 * 256];
            }
            acc = fmaf(w, s, acc);
        }
    }

    heads[rowq * 256 + m * 32 + lane] = acc;
}

// ---------------------------------------------------------------------------
extern "C" void kernel_launch(void* const* d_in, const int* in_sizes, int n_in,
                              void* d_out, int out_size, void* d_ws, size_t ws_size,
                              hipStream_t stream)
{
    const float* query  = (const float*)d_in[0];
    const float* ref    = (const float*)d_in[1];
    const float* value  = (const float*)d_in[2];
    // d_in[3] = value_spatial_shapes (static, baked into kernels)
    const float* W_val  = (const float*)d_in[4];
    const float* b_val  = (const float*)d_in[5];
    const float* W_off  = (const float*)d_in[6];
    const float* b_off  = (const float*)d_in[7];
    const float* W_attn = (const float*)d_in[8];
    const float* b_attn = (const float*)d_in[9];
    const float* W_out  = (const float*)d_in[10];
    const float* b_out  = (const float*)d_in[11];
    float* out = (float*)d_out;

    float* ws     = (float*)d_ws;
    float* v_proj = ws;                                 // ROWS*256
    float* off    = v_proj + (size_t)ROWS * 256;        // ROWS*256
    float* aw     = off    + (size_t)ROWS * 256;        // ROWS*128
    float* heads  = aw     + (size_t)ROWS * 128;        // ROWS*256

    const dim3 gblk(128);  // 4 waves -> 256 output columns per block

    // 1) value projection: v = value @ W_val + b_val     [43520x256x256]
    gemm_wmma_f32<<<dim3(ROWS / 16, 1), gblk, 0, stream>>>(
        value, W_val, b_val, v_proj, ROWS, 256);

    // 2) sampling offsets: off = query @ W_off + b_off   [43520x256x256]
    gemm_wmma_f32<<<dim3(ROWS / 16, 1), gblk, 0, stream>>>(
        query, W_off, b_off, off, ROWS, 256);

    // 3) attention logits: aw = query @ W_attn + b_attn  [43520x128x256]
    gemm_wmma_f32<<<dim3(ROWS / 16, 1), gblk, 0, stream>>>(
        query, W_attn, b_attn, aw, ROWS, 128);

    // 4) softmax over the 16 (level,point) logits per (n,q,head)
    {
        const int nrows = ROWS * NH;
        softmax16_kernel<<<(nrows + 255) / 256, 256, 0, stream>>>(aw, nrows);
    }

    // 5) bilinear deformable sampling -> heads [ROWS, 256]
    {
        const int nwaves = ROWS * NH;          // 348160 waves
        msda_sample_kernel<<<nwaves / 8, 256, 0, stream>>>(v_proj, off, aw, ref, heads);
    }

    // 6) output projection: out = heads @ W_out + b_out  [43520x256x256]
    gemm_wmma_f32<<<dim3(ROWS / 16, 1), gblk, 0, stream>>>(
        heads, W_out, b_out, out, ROWS, 256);
}